// LmulLinear_66048007078583
// MI455X (gfx1250) — compile-verified
//
#include <hip/hip_runtime.h>
#include <stdint.h>

// L-mul linear layer: out[i,j] = bias[j] + sum_k bitcast_f32(bits(x[i,k]) + bits(w[j,k]) - OFFSET)
// (uint32 wraparound add). The lmul core is a nonlinear bitwise op => not bilinear => no WMMA
// can reproduce it; it is VALU-bound (~134M v_add ops, ~1.8MB HBM traffic).
// CDNA5 paths used:
//   * GLOBAL_LOAD_ASYNC_TO_LDS_B128 double-buffered staging (ASYNCcnt + s_wait_asynccnt)
//   * wave32 VOPD-friendly (v_add_nc_u32 + v_add_f32) inner loop, ds_load_b128 fragments
//   * split-K x4 with deterministic workspace reduction
//   * V_WMMA_F32_16X16X4_F32 in the reduction epilogue: bias broadcast-add expressed as the
//     exact rank-1 matmul  D = ones(16x1) * bias(1x16) + C  (bit-identical to a single f32 add)

typedef __attribute__((ext_vector_type(4))) float        float4v;
typedef __attribute__((ext_vector_type(2))) float        float2v;
typedef __attribute__((ext_vector_type(8))) float        float8v;
typedef __attribute__((ext_vector_type(4))) unsigned int uint4v;

#define M_DIM 256
#define K_DIM 512
#define P_DIM 512
#define LMUL_OFFSET 1064828928u   // 0x3F780000

#define BM 64
#define BN 64
#define BK 32
#define KSLICE 128                 // split-K factor 4
#define LDS_STRIDE 36              // 32 + 4 pad; 36*4 = 144 B, multiple of 16 -> b128 ok
#define TILE_ELEMS (BM * LDS_STRIDE)

__global__ __launch_bounds__(256) void lmul_main_kernel(const float* __restrict__ x,
                                                        const float* __restrict__ w,
                                                        float* __restrict__ part) {
    __shared__ __align__(16) unsigned int smem[2][2][TILE_ELEMS];  // [buf][A/B][64x36]

    const int tid = threadIdx.x;
    const int j0  = blockIdx.x * BN;        // P tile
    const int i0  = blockIdx.y * BM;        // M tile
    const int ks0 = blockIdx.z * KSLICE;    // K slice

    const int tc = tid & 15;                // 16 column groups of 4
    const int tr = tid >> 4;                // 16 row groups of 4

    // Low 32 bits of the flat shared-aperture address are the LDS byte offset.
    const uint32_t lds_base = (uint32_t)(uintptr_t)&smem[0][0][0];

    // Stage one BMxBK (and BNxBK) tile pair with async b128 copies.
    // 64 rows x 8 float4 per operand = 512 float4; 256 threads -> 2 each.
    auto stage = [&](int buf, int ks) {
#pragma unroll
        for (int it = 0; it < 2; ++it) {
            const int q   = it * 256 + tid;   // 0..511
            const int row = q >> 3;           // 0..63
            const int k4  = q & 7;            // float4 index along k

            const float* ga = x + (size_t)(i0 + row) * K_DIM + ks + k4 * 4;
            const uint32_t la = lds_base +
                (uint32_t)(((buf * 2 + 0) * TILE_ELEMS + row * LDS_STRIDE + k4 * 4) * 4);
            asm volatile("global_load_async_to_lds_b128 %0, %1, off"
                         :: "v"(la), "v"(ga) : "memory");

            const float* gb = w + (size_t)(j0 + row) * K_DIM + ks + k4 * 4;
            const uint32_t lb = lds_base +
                (uint32_t)(((buf * 2 + 1) * TILE_ELEMS + row * LDS_STRIDE + k4 * 4) * 4);
            asm volatile("global_load_async_to_lds_b128 %0, %1, off"
                         :: "v"(lb), "v"(gb) : "memory");
        }
    };

    float4v acc[4][4] = {};

    stage(0, ks0);
    asm volatile("s_wait_asynccnt 0x0" ::: "memory");
    __syncthreads();

    int buf = 0;
    const int ksteps = KSLICE / BK;  // 4
    for (int s = 0; s < ksteps; ++s) {
        if (s + 1 < ksteps) stage(buf ^ 1, ks0 + (s + 1) * BK);  // overlap copy w/ compute

        const unsigned int* As = smem[buf][0];
        const unsigned int* Bs = smem[buf][1];
#pragma unroll
        for (int kk = 0; kk < BK; kk += 4) {
            uint4v af[4], bf[4];
#pragma unroll
            for (int r = 0; r < 4; ++r)
                af[r] = *(const uint4v*)&As[(tr * 4 + r) * LDS_STRIDE + kk];
#pragma unroll
            for (int c = 0; c < 4; ++c)  // fold -OFFSET into B fragment (amortized 4x)
                bf[c] = *(const uint4v*)&Bs[(tc * 4 + c) * LDS_STRIDE + kk] - LMUL_OFFSET;
#pragma unroll
            for (int r = 0; r < 4; ++r)
#pragma unroll
                for (int c = 0; c < 4; ++c) {
                    uint4v sbits = af[r] + bf[c];                       // v_add_nc_u32 x4
                    acc[r][c] += __builtin_bit_cast(float4v, sbits);    // v_add_f32 x4
                }
        }

        asm volatile("s_wait_asynccnt 0x0" ::: "memory");
        __syncthreads();
        buf ^= 1;
    }

    // Write this K-slice's partial tile (horizontal-reduce the 4-wide accumulators).
    float* out = part + (size_t)blockIdx.z * (M_DIM * P_DIM);
#pragma unroll
    for (int r = 0; r < 4; ++r) {
        float4v v;
#pragma unroll
        for (int c = 0; c < 4; ++c)
            v[c] = (acc[r][c][0] + acc[r][c][1]) + (acc[r][c][2] + acc[r][c][3]);
        *(float4v*)&out[(size_t)(i0 + tr * 4 + r) * P_DIM + j0 + tc * 4] = v;
    }
}

// Reduction epilogue: one wave per 16x16 output tile (2 tiles/wave).
// C fragment built directly in the WMMA f32 16x16 C/D VGPR layout:
//   VGPR v, lane l  ->  element (M = v + (l>=16 ? 8 : 0), N = l&15).
// Bias broadcast-add done on the matrix pipe: D = ones_col(A) x bias_row(B) + C (exact).
__global__ __launch_bounds__(256) void lmul_reduce_wmma_kernel(const float* __restrict__ part,
                                                               const float* __restrict__ bias,
                                                               float* __restrict__ out) {
    const int lane = threadIdx.x & 31;
    const int wave = blockIdx.x * (256 >> 5) + (threadIdx.x >> 5);  // 0..255
    const int ln   = lane & 15;       // N within tile
    const int lh   = lane >> 4;       // M offset 0 / 8

    // A (16x4): K=0 column = 1.0 (lanes 0-15 of VGPR0); K=1..3 = 0.
    float2v a;
    a[0] = (lane < 16) ? 1.0f : 0.0f;
    a[1] = 0.0f;

#pragma unroll
    for (int tt = 0; tt < 2; ++tt) {
        const int tile = wave * 2 + tt;          // 0..511
        const int i0   = (tile >> 5) * 16;       // 16 M tiles
        const int j0   = (tile & 31) * 16;       // 32 N tiles

        // C = sum of the 4 split-K partials, loaded in C-matrix layout.
        float8v c;
#pragma unroll
        for (int v = 0; v < 8; ++v) {
            const size_t off = (size_t)(i0 + v + lh * 8) * P_DIM + j0 + ln;
            float s = part[off];
#pragma unroll
            for (int z = 1; z < 4; ++z)
                s += part[(size_t)z * (M_DIM * P_DIM) + off];
            c[v] = s;
        }

        // B (4x16): K=0 row = bias[j0..j0+15]; other rows = 0.
        const float bv = bias[j0 + ln];
        float2v b;
        b[0] = (lane < 16) ? bv : 0.0f;
        b[1] = 0.0f;

        // D = A x B + C  ==  C[i,j] + bias[j]   (exact: 1.0*bias + zero products)
        const float8v d = __builtin_amdgcn_wmma_f32_16x16x4_f32(
            /*neg_a=*/false, a, /*neg_b=*/false, b,
            /*c_mod=*/(short)0, c, /*reuse_a=*/false, /*reuse_b=*/false);

#pragma unroll
        for (int v = 0; v < 8; ++v)
            out[(size_t)(i0 + v + lh * 8) * P_DIM + j0 + ln] = d[v];
    }
}

extern "C" void kernel_launch(void* const* d_in, const int* in_sizes, int n_in,
                              void* d_out, int out_size, void* d_ws, size_t ws_size,
                              hipStream_t stream) {
    (void)in_sizes; (void)n_in; (void)out_size; (void)ws_size;
    const float* x    = (const float*)d_in[0];   // (256, 512)
    const float* w    = (const float*)d_in[1];   // (512, 512) row-major: w[j*512 + k]
    const float* bias = (const float*)d_in[2];   // (512,)
    float* ws  = (float*)d_ws;                   // 4 * 256*512 floats = 2 MiB partials
    float* out = (float*)d_out;                  // (256, 512)

    dim3 grid(P_DIM / BN, M_DIM / BM, K_DIM / KSLICE);  // (8, 4, 4) = 128 blocks
    lmul_main_kernel<<<grid, 256, 0, stream>>>(x, w, ws);

    // 512 tiles of 16x16, 2 per wave, 8 waves per block -> 32 blocks.
    lmul_reduce_wmma_kernel<<<32, 256, 0, stream>>>(ws, bias, out);
}